// Net_1236950581524
// MI455X (gfx1250) — compile-verified
//
#include <hip/hip_runtime.h>
#include <math.h>

typedef __attribute__((ext_vector_type(2))) float v2f;
typedef __attribute__((ext_vector_type(8))) float v8f;

#define BATCH   8
#define NPTS    4096
#define ROWS    16                  // query rows per block (== waves per block)
#define WAVES   16
#define THREADS1 512
#define KSEL    300
#define SORTN   512
#define JT_PER_WAVE ((NPTS / 16) / WAVES)   // 16 column tiles per wave

#define DIST_FLOATS (ROWS * NPTS)
#define LDS_BYTES (DIST_FLOATS * 4 + 16 * 4 + WAVES * 256 * 4 + WAVES * SORTN * 4)

// ---------------------------------------------------------------------------
// Kernel 1: per (batch, 16-row tile): WMMA f32 16x16x4 distance tile kept in
// LDS, then per-row exact radix-select of the 300 nearest + bitonic sort of
// the survivors. Writes rank-ordered neighbor indices to workspace.
// ---------------------------------------------------------------------------
__global__ __launch_bounds__(THREADS1) void topk_kernel(const float* __restrict__ pts,
                                                        int* __restrict__ nbr) {
  extern __shared__ char smem[];
  float*    dist    = (float*)smem;                       // ROWS x NPTS
  float*    sqi     = dist + DIST_FLOATS;                 // 16
  unsigned* histAll = (unsigned*)(sqi + 16);              // WAVES x 256
  int*      selAll  = (int*)(histAll + WAVES * 256);      // WAVES x 512

  const int b    = blockIdx.x / (NPTS / ROWS);
  const int tile = blockIdx.x % (NPTS / ROWS);
  const int i0   = tile * ROWS;
  const int lane = threadIdx.x & 31;
  const int wave = threadIdx.x >> 5;
  const int half = lane >> 4;
  const int lm   = lane & 15;
  const float* P = pts + (size_t)b * NPTS * 3;

  // |p_i|^2 for the 16 query rows
  if (threadIdx.x < ROWS) {
    const float x = P[(i0 + threadIdx.x) * 3 + 0];
    const float y = P[(i0 + threadIdx.x) * 3 + 1];
    const float z = P[(i0 + threadIdx.x) * 3 + 2];
    sqi[threadIdx.x] = x * x + y * y + z * z;
  }
  __syncthreads();

  // A fragment (16x4 f32): row m = i0+lm; half 0 holds (x,y), half 1 holds (z,0)
  v2f afrag;
  {
    const float* q = P + (size_t)(i0 + lm) * 3;
    if (half == 0) { afrag.x = q[0]; afrag.y = q[1]; }
    else           { afrag.x = q[2]; afrag.y = 0.0f; }
  }
  float si[8];
#pragma unroll
  for (int v = 0; v < 8; ++v) si[v] = sqi[half * 8 + v];

  // Phase 1: 16 waves cover 256 column tiles; dist tile stays in LDS.
  for (int t = 0; t < JT_PER_WAVE; ++t) {
    const int j0 = (t * WAVES + wave) * 16;
    const float* kp = P + (size_t)(j0 + lm) * 3;
    const float kx = kp[0], ky = kp[1], kz = kp[2];
    const float sj = kx * kx + ky * ky + kz * kz;
    v2f bfrag;
    if (half == 0) { bfrag.x = kx; bfrag.y = ky; }
    else           { bfrag.x = kz; bfrag.y = 0.0f; }
    v8f c = {0.f, 0.f, 0.f, 0.f, 0.f, 0.f, 0.f, 0.f};
    v8f d = __builtin_amdgcn_wmma_f32_16x16x4_f32(false, afrag, false, bfrag,
                                                  (short)0, c, false, false);
#pragma unroll
    for (int v = 0; v < 8; ++v) {
      float d2 = si[v] + sj - 2.0f * d[v];
      d2 = fmaxf(d2, 0.0f);
      dist[(v + 8 * half) * NPTS + j0 + lm] = d2;
    }
  }
  __syncthreads();

  // Phase 2: one wave per row. Radix-select 300th-smallest over fp32 bits.
  const int r = wave;
  float*    drow = dist + r * NPTS;
  unsigned* h    = histAll + wave * 256;
  int*      sel  = selAll + wave * SORTN;

  unsigned prefix = 0;
  int want = KSEL;       // 1-indexed rank we are hunting
  int lessTotal = 0;     // elements strictly below final threshold
  for (int p = 3; p >= 0; --p) {
    for (int i = lane; i < 256; i += 32) h[i] = 0u;
    __syncthreads();
    for (int tt = lane; tt < NPTS; tt += 32) {
      const unsigned u = __float_as_uint(drow[tt]);
      const bool ok = (p == 3) || ((u >> ((p + 1) * 8)) == prefix);
      if (ok) atomicAdd(&h[(u >> (p * 8)) & 255u], 1u);
    }
    __syncthreads();
    int binSel = 0, below = 0;
    if (lane == 0) {
      int cum = 0;
      for (int bin = 0; bin < 256; ++bin) {
        const int cnt = (int)h[bin];
        if (cum + cnt >= want) { binSel = bin; below = cum; break; }
        cum += cnt;
      }
    }
    binSel = __shfl(binSel, 0);
    below  = __shfl(below, 0);
    prefix = (prefix << 8) | (unsigned)binSel;
    want -= below;
    lessTotal += below;
    __syncthreads();
  }
  const unsigned Tbits = prefix;       // bit pattern of the 300th smallest d2
  const int tiesNeeded = want;         // = KSEL - lessTotal

  // Deterministic compaction: ballot-prefix ordering == ascending index order.
  {
    int lessRun = 0, tieRun = 0;
    const unsigned lowMask = (1u << lane) - 1u;
    for (int base = 0; base < NPTS; base += 32) {
      const int tt = base + lane;
      const unsigned u = __float_as_uint(drow[tt]);
      const bool pl = (u < Tbits);
      const bool pe = (u == Tbits);
      const unsigned mL = (unsigned)__ballot(pl);
      const unsigned mE = (unsigned)__ballot(pe);
      if (pl) sel[lessRun + __popc(mL & lowMask)] = tt;
      const int et = tieRun + __popc(mE & lowMask);
      if (pe && et < tiesNeeded) sel[lessTotal + et] = tt;
      lessRun += __popc(mL);
      tieRun  += __popc(mE);
    }
    for (int tt = KSEL + lane; tt < SORTN; tt += 32) sel[tt] = -1;
  }
  __syncthreads();

  // Bitonic sort of 512 entries keyed on (d2 bits, idx); sentinels sort last.
  for (int k = 2; k <= SORTN; k <<= 1) {
    for (int j = k >> 1; j > 0; j >>= 1) {
      for (int tt = lane; tt < SORTN; tt += 32) {
        const int ixj = tt ^ j;
        if (ixj > tt) {
          const int ia = sel[tt];
          const int ib = sel[ixj];
          const unsigned long long ka = (ia < 0) ? ~0ull
              : ((((unsigned long long)__float_as_uint(drow[ia])) << 32) | (unsigned)ia);
          const unsigned long long kb = (ib < 0) ? ~0ull
              : ((((unsigned long long)__float_as_uint(drow[ib])) << 32) | (unsigned)ib);
          const bool up = ((tt & k) == 0);
          if ((ka > kb) == up) { sel[tt] = ib; sel[ixj] = ia; }
        }
      }
      __syncthreads();
    }
  }

  int* onbr = nbr + ((size_t)b * NPTS + (i0 + r)) * KSEL;
  for (int tt = lane; tt < KSEL; tt += 32) onbr[tt] = sel[tt];
}

// ---------------------------------------------------------------------------
// Kernel 2: PPF features. One wave per query point; lanes stride each rank
// bucket; __shfl_xor reduction; lane 0 writes the bucket mean.
// ---------------------------------------------------------------------------
__device__ __forceinline__ float ang3(float ax, float ay, float az,
                                      float bx, float by, float bz) {
  const float y  = ax * bx + ay * by + az * bz;
  const float cx = ay * bz - az * by;
  const float cy = az * bx - ax * bz;
  const float cz = ax * by - ay * bx;
  const float x  = sqrtf(cx * cx + cy * cy + cz * cz);
  return atan2f(x, y) * 0.31830988618379067f;   // 1/pi
}

__global__ __launch_bounds__(256) void ppf_kernel(const float* __restrict__ pts,
                                                  const float* __restrict__ nrm,
                                                  const int* __restrict__ nbr,
                                                  float* __restrict__ out) {
  const int lane = threadIdx.x & 31;
  const int wave = threadIdx.x >> 5;
  const int p = blockIdx.x * 8 + wave;         // global point id
  const int b = p / NPTS;
  const int i = p % NPTS;
  const float* P  = pts + (size_t)b * NPTS * 3;
  const float* Nm = nrm + (size_t)b * NPTS * 3;
  const float pix = P[i * 3 + 0], piy = P[i * 3 + 1], piz = P[i * 3 + 2];
  const float nix = Nm[i * 3 + 0], niy = Nm[i * 3 + 1], niz = Nm[i * 3 + 2];
  const int* nb = nbr + (size_t)p * KSEL;
  const int bounds[7] = {0, 10, 20, 40, 80, 160, 300};
  float* o = out + (size_t)p * 24;

#pragma unroll
  for (int g = 0; g < 6; ++g) {
    float s0 = 0.f, s1 = 0.f, s2 = 0.f, s3 = 0.f;
    for (int t = bounds[g] + lane; t < bounds[g + 1]; t += 32) {
      const int j = nb[t];
      const float dx = P[j * 3 + 0] - pix;
      const float dy = P[j * 3 + 1] - piy;
      const float dz = P[j * 3 + 2] - piz;
      const float njx = Nm[j * 3 + 0], njy = Nm[j * 3 + 1], njz = Nm[j * 3 + 2];
      s0 += sqrtf(dx * dx + dy * dy + dz * dz);
      s1 += ang3(nix, niy, niz, dx, dy, dz);
      s2 += ang3(njx, njy, njz, dx, dy, dz);
      s3 += ang3(nix, niy, niz, njx, njy, njz);
    }
#pragma unroll
    for (int off = 16; off >= 1; off >>= 1) {
      s0 += __shfl_xor(s0, off);
      s1 += __shfl_xor(s1, off);
      s2 += __shfl_xor(s2, off);
      s3 += __shfl_xor(s3, off);
    }
    if (lane == 0) {
      const float inv = 1.0f / (float)(bounds[g + 1] - bounds[g]);
      o[g * 4 + 0] = s0 * inv;
      o[g * 4 + 1] = s1 * inv;
      o[g * 4 + 2] = s2 * inv;
      o[g * 4 + 3] = s3 * inv;
    }
  }
}

extern "C" void kernel_launch(void* const* d_in, const int* in_sizes, int n_in,
                              void* d_out, int out_size, void* d_ws, size_t ws_size,
                              hipStream_t stream) {
  (void)in_sizes; (void)n_in; (void)out_size; (void)ws_size;
  const float* pts = (const float*)d_in[0];
  const float* nrm = (const float*)d_in[1];
  float* out = (float*)d_out;
  int*   nbr = (int*)d_ws;    // 8*4096*300 int32 = 39.3 MB of workspace

  // 311,360 B of dynamic LDS (> default cap) — raise the limit (idempotent).
  (void)hipFuncSetAttribute((const void*)topk_kernel,
                            hipFuncAttributeMaxDynamicSharedMemorySize, LDS_BYTES);

  topk_kernel<<<BATCH * (NPTS / ROWS), THREADS1, LDS_BYTES, stream>>>(pts, nbr);
  ppf_kernel<<<(BATCH * NPTS) / 8, 256, 0, stream>>>(pts, nrm, nbr, out);
}